// NodeDenoisingADMM_82197084110902
// MI455X (gfx1250) — compile-verified
//
#include <hip/hip_runtime.h>

// NodeDenoisingADMM on MI455X (gfx1250, wave32).
// L2-resident scatter/gather problem: 28 edge-parallel SpMM passes (atomic
// f32 segment-sum in L2) + fused elementwise ADMM updates. CDNA5 path:
// double-buffered GLOBAL_LOAD_ASYNC_TO_LDS_B64 gather + s_wait_asynccnt.

#define NN     20000
#define FEAT   64
#define KM     4
#define NNZ_   320000
#define EDGES  (KM * NNZ_)
#define GAMMA_ 1.0f
#define NITER  14

#ifndef __has_builtin
#define __has_builtin(x) 0
#endif

#if defined(__HIP_DEVICE_COMPILE__) && __has_builtin(__builtin_amdgcn_global_load_async_to_lds_b64)
#define USE_ASYNC 1
#else
#define USE_ASYNC 0
#endif

#if USE_ASYNC
// 64-bit payload type per the builtin's signature (clang diagnostic:
// parameter type is 'int __attribute__((vector_size(2*sizeof(int))))*').
typedef int v2i_ __attribute__((vector_size(8)));
typedef __attribute__((address_space(3))) v2i_* lds_v2i_p;

__device__ __forceinline__ void async_gather_b64(const float* src, float* lds_dst) {
  __builtin_amdgcn_global_load_async_to_lds_b64(
      (v2i_*)src, (lds_v2i_p)lds_dst, 0, 0);
}
#endif

__device__ __forceinline__ float softthr(float x, float t) {
  return fmaxf(x - t, 0.0f) - fmaxf(-x - t, 0.0f);
}

__device__ __forceinline__ void atomAddF(float* p, float v) {
  // non-returning agent-scope f32 add -> global_atomic_add_f32 (no TH=RET)
  __hip_atomic_fetch_add(p, v, __ATOMIC_RELAXED, __HIP_MEMORY_SCOPE_AGENT);
}

// ---------------------------------------------------------------------------
// Precompute: Uk = F, num_f = d*F*mask^2, denom = d*mask^2 + gamma,
// thr[k][n] = (nu_k/gamma)*d[n], nu = [0, 2, 0.5, 0.125].
// ---------------------------------------------------------------------------
__global__ void __launch_bounds__(256)
k_pre(const float* __restrict__ F, const float* __restrict__ d,
      const float* __restrict__ mask, float* __restrict__ Uk,
      float* __restrict__ numf, float* __restrict__ den,
      float* __restrict__ thr) {
  int i = blockIdx.x * blockDim.x + threadIdx.x;
  if (i >= NN * FEAT) return;
  int n = i >> 6;
  float dn = d[n];
  float m  = mask[n];
  float m2 = m * m;
  float f  = F[i];
  Uk[i]   = f;
  numf[i] = dn * f * m2;
  if ((i & 63) == 0) {
    den[n]          = dn * m2 + GAMMA_;
    thr[n]          = 0.0f;
    thr[NN + n]     = 2.0f   * dn;
    thr[2 * NN + n] = 0.5f   * dn;
    thr[3 * NN + n] = 0.125f * dn;
  }
}

// ---------------------------------------------------------------------------
// Fused Y-update + soft-threshold + v-update over [K][N][FEAT] (float4):
//   Y  = first ? 0 : v + gamma*WU        (Y_{t} = v_{t-1} + gamma*WU_t)
//   Z  = soft(WU + Y/gamma, thr[k][n])
//   v  = Y - gamma*Z                     (written in place)
// Z and Y never hit memory.
// ---------------------------------------------------------------------------
__global__ void __launch_bounds__(256)
k_fuseA(const float* __restrict__ WU, float* __restrict__ v,
        const float* __restrict__ thr, int first) {
  int i4 = blockIdx.x * blockDim.x + threadIdx.x;
  const int total4 = KM * NN * FEAT / 4;
  if (i4 >= total4) return;
  int kn = (i4 * 4) / FEAT;  // == k*NN + n (layout [k][n][f])
  float t = thr[kn];
  float4 wu = ((const float4*)WU)[i4];
  float4 y;
  if (first) {
    y = make_float4(0.f, 0.f, 0.f, 0.f);
  } else {
    float4 vv = ((const float4*)v)[i4];
    y.x = vv.x + GAMMA_ * wu.x;
    y.y = vv.y + GAMMA_ * wu.y;
    y.z = vv.z + GAMMA_ * wu.z;
    y.w = vv.w + GAMMA_ * wu.w;
  }
  const float ig = 1.0f / GAMMA_;
  float4 o;
  o.x = y.x - GAMMA_ * softthr(wu.x + y.x * ig, t);
  o.y = y.y - GAMMA_ * softthr(wu.y + y.y * ig, t);
  o.z = y.z - GAMMA_ * softthr(wu.z + y.z * ig, t);
  o.w = y.w - GAMMA_ * softthr(wu.w + y.w * ig, t);
  ((float4*)v)[i4] = o;
}

// ---------------------------------------------------------------------------
// Uk = (num_f - WTV) / denom   over [N][FEAT] (float4)
// ---------------------------------------------------------------------------
__global__ void __launch_bounds__(256)
k_uk(const float* __restrict__ numf, const float* __restrict__ WTV,
     const float* __restrict__ den, float* __restrict__ Uk) {
  int i4 = blockIdx.x * blockDim.x + threadIdx.x;
  const int total4 = NN * FEAT / 4;
  if (i4 >= total4) return;
  int n = (i4 * 4) >> 6;
  float dd = den[n];
  float4 a = ((const float4*)numf)[i4];
  float4 b = ((const float4*)WTV)[i4];
  float4 o = make_float4((a.x - b.x) / dd, (a.y - b.y) / dd,
                         (a.z - b.z) / dd, (a.w - b.w) / dd);
  ((float4*)Uk)[i4] = o;
}

// ---------------------------------------------------------------------------
// Edge-parallel SpMM, one wave32 per edge, 2 floats/lane.
//   mode 0: WU[k*NN + rows[e]] += vals[e] * X[cols[e]]          (X = Uk, [N][F])
//   mode 1: WTV[rows[e]]       += vals[e] * X[k*NN + cols[e]]   (X = v, [K][N][F])
// Async path: per-wave double-buffered global->LDS gather of the 256B feature
// row of the NEXT edge while the current one is multiplied & scattered.
// OUT must be pre-zeroed.
// ---------------------------------------------------------------------------
__global__ void __launch_bounds__(256)
k_spmm(const int* __restrict__ rows, const int* __restrict__ cols,
       const float* __restrict__ vals, const float* __restrict__ X,
       float* __restrict__ OUT, int mode) {
  const int lane = threadIdx.x & 31;
  const int wIn  = threadIdx.x >> 5;
  const int wave = (blockIdx.x * blockDim.x + threadIdx.x) >> 5;
  const int nW   = (gridDim.x * blockDim.x) >> 5;

#if USE_ASYNC
  __shared__ float lbuf[8][2][FEAT];  // 8 waves x 2 buffers x 64 floats = 4KB
  int e = wave;
  int cur = 0;
  int r = 0, k = 0;
  float w = 0.0f;
  if (e < EDGES) {  // e is wave-uniform: whole wave takes same path
    r = rows[e]; k = e / NNZ_; w = vals[e];
    int c  = cols[e];
    long g = mode ? ((long)k * NN + c) : (long)c;
    async_gather_b64(X + g * FEAT + lane * 2, &lbuf[wIn][0][lane * 2]);
  }
  while (e < EDGES) {
    const int e1 = e + nW;
    int rn = 0, kn = 0;
    float wn = 0.0f;
    const bool pn = (e1 < EDGES);
    if (pn) {
      rn = rows[e1]; kn = e1 / NNZ_; wn = vals[e1];
      int cn  = cols[e1];
      long gn = mode ? ((long)kn * NN + cn) : (long)cn;
      async_gather_b64(X + gn * FEAT + lane * 2, &lbuf[wIn][cur ^ 1][lane * 2]);
      // async loads complete in order: <=1 outstanding => current buffer ready
      asm volatile("s_wait_asynccnt 1" ::: "memory");
    } else {
      asm volatile("s_wait_asynccnt 0" ::: "memory");
    }
    float x0 = lbuf[wIn][cur][lane * 2];
    float x1 = lbuf[wIn][cur][lane * 2 + 1];
    long s   = mode ? (long)r : ((long)k * NN + r);
    float* o = OUT + s * FEAT + lane * 2;
    atomAddF(o,     w * x0);
    atomAddF(o + 1, w * x1);
    r = rn; k = kn; w = wn; cur ^= 1; e = e1;
  }
#else
  for (long e = wave; e < EDGES; e += nW) {
    int r = rows[e], c = cols[e];
    float w = vals[e];
    int k = (int)(e / NNZ_);
    long g = mode ? ((long)k * NN + c) : (long)c;
    const float2 x = *(const float2*)(X + g * FEAT + lane * 2);
    long s   = mode ? (long)r : ((long)k * NN + r);
    float* o = OUT + s * FEAT + lane * 2;
    atomAddF(o,     w * x.x);
    atomAddF(o + 1, w * x.y);
  }
#endif
}

// ---------------------------------------------------------------------------
extern "C" void kernel_launch(void* const* d_in, const int* in_sizes, int n_in,
                              void* d_out, int out_size, void* d_ws, size_t ws_size,
                              hipStream_t stream) {
  (void)in_sizes; (void)n_in; (void)out_size; (void)ws_size;
  const float* F    = (const float*)d_in[0];
  const int*   rows = (const int*)d_in[1];
  const int*   cols = (const int*)d_in[2];
  const float* vals = (const float*)d_in[3];
  const float* dvec = (const float*)d_in[4];
  const float* mask = (const float*)d_in[5];
  // d_in[6] = thres_iter (constant 15 -> 14 scan steps)

  float* Uk = (float*)d_out;  // [N][FEAT], final value is the output
  float* ws = (float*)d_ws;
  size_t off = 0;
  float* v    = ws + off; off += (size_t)KM * NN * FEAT;  // 5.12M
  float* WU   = ws + off; off += (size_t)KM * NN * FEAT;  // 5.12M
  float* WTV  = ws + off; off += (size_t)NN * FEAT;       // 1.28M
  float* numf = ws + off; off += (size_t)NN * FEAT;       // 1.28M
  float* den  = ws + off; off += NN;
  float* thr  = ws + off; off += (size_t)KM * NN;
  // total ~51.6 MB of f32 scratch

  const int tpb  = 256;
  const int gPre = (NN * FEAT + tpb - 1) / tpb;
  const int gEw  = (KM * NN * FEAT / 4 + tpb - 1) / tpb;
  const int gUk  = (NN * FEAT / 4 + tpb - 1) / tpb;
  const int gSp  = 4000;  // 32000 waves, ~40 edges each

  k_pre<<<gPre, tpb, 0, stream>>>(F, dvec, mask, Uk, numf, den, thr);

  // WU = spmm_shared(Uk = F)
  (void)hipMemsetAsync(WU, 0, (size_t)KM * NN * FEAT * sizeof(float), stream);
  k_spmm<<<gSp, tpb, 0, stream>>>(rows, cols, vals, Uk, WU, 0);

  for (int it = 0; it < NITER; ++it) {
    // Y = v + g*WU ; Z = soft(WU + Y/g, thr) ; v = Y - g*Z
    k_fuseA<<<gEw, tpb, 0, stream>>>(WU, v, thr, it == 0 ? 1 : 0);
    // WTV = sum_k W_k v_k
    (void)hipMemsetAsync(WTV, 0, (size_t)NN * FEAT * sizeof(float), stream);
    k_spmm<<<gSp, tpb, 0, stream>>>(rows, cols, vals, v, WTV, 1);
    // Uk = (num_f - WTV) / denom
    k_uk<<<gUk, tpb, 0, stream>>>(numf, WTV, den, Uk);
    // WU = spmm_shared(Uk)  (cached: reused by next iter's Y update AND Z)
    if (it + 1 < NITER) {
      (void)hipMemsetAsync(WU, 0, (size_t)KM * NN * FEAT * sizeof(float), stream);
      k_spmm<<<gSp, tpb, 0, stream>>>(rows, cols, vals, Uk, WU, 0);
    }
  }
}